// CTCLossLayer_27496380629483
// MI455X (gfx1250) — compile-verified
//
#include <hip/hip_runtime.h>
#include <hip/hip_bf16.h>

#define CTC_B 128
#define CTC_T 1000
#define CTC_V 96
#define CTC_L 64
#define CTC_S 129            // 2*L+1
#define NEGV (-1e30f)

#if defined(__has_builtin)
#  if __has_builtin(__builtin_amdgcn_global_load_async_to_lds_b32)
#    define HAS_ASYNC 1
#  else
#    define HAS_ASYNC 0
#  endif
#else
#  define HAS_ASYNC 0
#endif

// Raw SOPP waits (memory-clobber so the compiler cannot move LDS ops across them).
#define WAIT_ASYNC3() asm volatile("s_wait_asynccnt 3" ::: "memory")
#define WAIT_ASYNC0() asm volatile("s_wait_asynccnt 0" ::: "memory")
#define WAIT_DS0()    asm volatile("s_wait_dscnt 0"    ::: "memory")

__device__ __forceinline__ float wave_sum_f(float x) {
#pragma unroll
  for (int off = 16; off; off >>= 1) x += __shfl_xor(x, off, 32);
  return x;
}
__device__ __forceinline__ float wave_max_f(float x) {
#pragma unroll
  for (int off = 16; off; off >>= 1) x = fmaxf(x, __shfl_xor(x, off, 32));
  return x;
}
__device__ __forceinline__ int wave_sum_i(int x) {
#pragma unroll
  for (int off = 16; off; off >>= 1) x += __shfl_xor(x, off, 32);
  return x;
}

// log(e^a + e^b + e^c), stable; NEGV acts as log-domain zero.
__device__ __forceinline__ float lae3(float a, float b, float c) {
  float m = fmaxf(a, fmaxf(b, c));
  float s = __expf(a - m) + __expf(b - m) + __expf(c - m);
  return m + __logf(s);
}

// ---------------------------------------------------------------------------
// Kernel 1: per-(b,t) logsumexp over V=96, one wave per row. Fully parallel.
// ---------------------------------------------------------------------------
__global__ __launch_bounds__(256) void k_lse(const float* __restrict__ yp,
                                             float* __restrict__ lse) {
  int w = (int)((blockIdx.x * 256 + threadIdx.x) >> 5);   // row id = b*T + t
  int l = (int)(threadIdx.x & 31);
  if (w >= CTC_B * CTC_T) return;
  const float* row = yp + (size_t)w * CTC_V;
  float v0 = row[l], v1 = row[l + 32], v2 = row[l + 64];
  float m = wave_max_f(fmaxf(v0, fmaxf(v1, v2)));
  float s = wave_sum_f(__expf(v0 - m) + __expf(v1 - m) + __expf(v2 - m));
  if (l == 0) lse[w] = m + __logf(s);
}

// ---------------------------------------------------------------------------
// Kernel 2: serial CTC scan, one wave32 per batch element.
// Raw-logit recursion (shift-equivariance); sum(lse) subtracted at the end.
// Logit rows double-buffered in LDS via CDNA5 async-to-LDS loads.
// ---------------------------------------------------------------------------
__global__ __launch_bounds__(32) void k_chain(const float* __restrict__ yp,
                                              const int* __restrict__ yt,
                                              const float* __restrict__ lse,
                                              float* __restrict__ lossv) {
  __shared__ float rb[2][128];         // two 96-float rows (padded)
  __shared__ float alpha[CTC_S + 7];   // 129 states (padded)

  const int b = (int)blockIdx.x;
  const int l = (int)threadIdx.x;
  const float* base = yp + (size_t)b * CTC_T * CTC_V;
  const int* lab = yt + b * CTC_L;

  // Static per-state metadata. Lane l owns s = l+32k (k=0..3) and s=128 (all
  // lanes compute state 128 redundantly -> uniform control flow, same value).
  int st[5], ei[5];
  bool sk[5];
#pragma unroll
  for (int k = 0; k < 5; ++k) {
    int s = l + 32 * k;
    if (s > 128) s = 128;
    st[k] = s;
    int e = (s & 1) ? lab[s >> 1] : 0;   // ext[s]: blank on even s
    ei[k] = e;
    bool skip = (s & 1) && ((s == 1) ? true : (e != lab[(s >> 1) - 1]));
    sk[k] = skip && (s >= 2);
  }

  // label length (count of nonzeros; == 64 for this input distribution)
  int cnt = (lab[l] != 0) + (lab[l + 32] != 0);
  cnt = wave_sum_i(cnt);

  // sum of lse over time (removed from the serial chain by shift-equivariance)
  float lsum = 0.f;
  {
    const float* lb = lse + b * CTC_T;
    for (int t = l; t < CTC_T; t += 32) lsum += lb[t];
    lsum = wave_sum_f(lsum);
  }

#if HAS_ASYNC
  auto issue_row = [&](int tt, int bufi) {
    const float* sp = base + (size_t)tt * CTC_V;
    float* dp = &rb[bufi][0];
    __builtin_amdgcn_global_load_async_to_lds_b32((int*)(sp + l),      (int*)(dp + l),      0, 0);
    __builtin_amdgcn_global_load_async_to_lds_b32((int*)(sp + l + 32), (int*)(dp + l + 32), 0, 0);
    __builtin_amdgcn_global_load_async_to_lds_b32((int*)(sp + l + 64), (int*)(dp + l + 64), 0, 0);
  };
  issue_row(0, 0);
  issue_row(1, 1);
  WAIT_ASYNC3();                 // row 0 resident (async loads retire in order)
#else
  { const float* sp = base;
    rb[0][l] = sp[l]; rb[0][l + 32] = sp[l + 32]; rb[0][l + 64] = sp[l + 64]; }
  __syncthreads();
#endif

  // alpha init at t=0 (raw logits)
#pragma unroll
  for (int k = 0; k < 5; ++k) alpha[st[k]] = NEGV;
  __syncthreads();
  if (l == 0) alpha[0] = rb[0][0];          // blank
  if (l == 1) alpha[1] = rb[0][ei[0]];      // first label (lane1 k0 has s=1)
  __syncthreads();

  for (int t = 1; t < CTC_T; ++t) {
#if HAS_ASYNC
    float* r = rb[t & 1];
    if (t < CTC_T - 1) { WAIT_ASYNC3(); } else { WAIT_ASYNC0(); }  // row t resident
#else
    float* r = rb[0];
    { const float* sp = base + (size_t)t * CTC_V;
      r[l] = sp[l]; r[l + 32] = sp[l + 32]; r[l + 64] = sp[l + 64]; }
    __syncthreads();
#endif

    float g[5], a[5], p[5], q[5];
#pragma unroll
    for (int k = 0; k < 5; ++k) {
      int s = st[k];
      g[k] = r[ei[k]];                       // gather raw logit for ext[s]
      a[k] = alpha[s];
      p[k] = (s > 0) ? alpha[s - 1] : NEGV;
      q[k] = sk[k] ? alpha[s - 2] : NEGV;
    }

#if HAS_ASYNC
    WAIT_DS0();                              // gathers landed before overwrite
    if (t + 2 < CTC_T) issue_row(t + 2, t & 1);
#endif

    float n[5];
#pragma unroll
    for (int k = 0; k < 5; ++k) n[k] = lae3(a[k], p[k], q[k]) + g[k];

    __syncthreads();                         // all reads done before writes
#pragma unroll
    for (int k = 0; k < 5; ++k) alpha[st[k]] = n[k];
    __syncthreads();                         // writes visible for next step
  }

  // terminal states: 2*len (trailing blank) and 2*len-1 (last label)
  int len = cnt;
  int ihi = 2 * len; if (ihi > 128) ihi = 128;
  int ilo = 2 * len - 1; if (ilo < 0) ilo = 0;
  float ah = alpha[ihi], al = alpha[ilo];
  float m = fmaxf(ah, al);
  float tot = m + __logf(__expf(ah - m) + __expf(al - m));
  if (l == 0) lossv[b] = -(tot - lsum);
}

// ---------------------------------------------------------------------------
// Kernel 3: mean over batch -> scalar output (deterministic tree reduce).
// ---------------------------------------------------------------------------
__global__ __launch_bounds__(32) void k_final(const float* __restrict__ lossv,
                                              float* __restrict__ out) {
  int l = (int)threadIdx.x;
  float x = lossv[l] + lossv[l + 32] + lossv[l + 64] + lossv[l + 96];
  x = wave_sum_f(x);
  if (l == 0) out[0] = x * (1.0f / (float)CTC_B);
}

extern "C" void kernel_launch(void* const* d_in, const int* in_sizes, int n_in,
                              void* d_out, int out_size, void* d_ws, size_t ws_size,
                              hipStream_t stream) {
  (void)in_sizes; (void)n_in; (void)out_size; (void)ws_size;
  const float* yp = (const float*)d_in[0];
  const int*   yt = (const int*)d_in[1];
  float* lse   = (float*)d_ws;                 // B*T floats  (512 KB)
  float* lossv = lse + CTC_B * CTC_T;          // B floats
  k_lse  <<<(CTC_B * CTC_T) / 8, 256, 0, stream>>>(yp, lse);
  k_chain<<<CTC_B, 32, 0, stream>>>(yp, yt, lse, lossv);
  k_final<<<1, 32, 0, stream>>>(lossv, (float*)d_out);
}